// StructuredTokenPruner_88682484728553
// MI455X (gfx1250) — compile-verified
//
#include <hip/hip_runtime.h>

// ---- fixed problem shape from the reference ----
#define BDIM 4
#define TDIM 8
#define CDIM 256
#define HWDIM 1024          // 32*32
#define KEEP  512           // keep_ratio 0.5 of 1024
#define NBT   (BDIM * TDIM) // 32
#define NPRUNED ((size_t)NBT * CDIM * HWDIM) // 8388608

typedef float v2f   __attribute__((ext_vector_type(2)));
typedef float v8f   __attribute__((ext_vector_type(8)));
typedef float f32x4 __attribute__((ext_vector_type(4)));

// ---------------------------------------------------------------------------
// Kernel A: per (b,t) compute token scores with V_WMMA_F32_16X16X4_F32,
// then exact top-k membership (jax tie-break: lower index first) -> mask.
// One block (1024 threads = 32 waves) per (b,t). Each wave owns 2 tiles of
// 16 positions; 64 WMMA K-steps cover the 256 channels.
// ---------------------------------------------------------------------------
__global__ __launch_bounds__(1024)
void tokenprune_score_select(const float* __restrict__ x,
                             float* __restrict__ mask_out)
{
    __shared__ float s_scores[HWDIM];

    const int bt = blockIdx.x;                     // 0..31
    const float* xbt = x + (size_t)bt * CDIM * HWDIM;

    const int tid  = threadIdx.x;
    const int wave = tid >> 5;                     // 0..31
    const int lane = tid & 31;
    const int half = lane >> 4;                    // 0/1: which half-wave
    const int l    = lane & 15;

    // B matrix = all ones (4x16 f32 -> 2 VGPRs/lane); layout irrelevant.
    v2f onesB; onesB.x = 1.0f; onesB.y = 1.0f;

    for (int tile = wave; tile < HWDIM / 16; tile += 32) {
        const int p0  = tile * 16;
        const int pos = p0 + l;                    // M = l for this half-wave
        // 32-bit A 16x4 layout: lane<16: VGPR0=K0, VGPR1=K1;
        //                       lane>=16: VGPR0=K2, VGPR1=K3.
        const float* base = xbt + (size_t)(half * 2) * HWDIM + pos;

        v8f acc = {};
        #pragma unroll 4
        for (int k = 0; k < CDIM; k += 4) {
            v2f a;
            a.x = __builtin_fabsf(base[(size_t)k * HWDIM]);          // chan k+2*half
            a.y = __builtin_fabsf(base[(size_t)k * HWDIM + HWDIM]);  // chan k+2*half+1
            // D = A x B + C : every column of D = sum over 4 channels
            acc = __builtin_amdgcn_wmma_f32_16x16x4_f32(
                      /*neg_a=*/false, a, /*neg_b=*/false, onesB,
                      /*c_mod=*/(short)0, acc, /*reuse_a=*/false, /*reuse_b=*/false);
        }

        // C/D layout: lanes 0-15 VGPR r -> M=r; lanes 16-31 VGPR r -> M=8+r.
        // Lane 0 extracts rows 0..7, lane 16 extracts rows 8..15.
        if (l == 0) {
            const int m0 = p0 + half * 8;
            #pragma unroll
            for (int r = 0; r < 8; ++r)
                s_scores[m0 + r] = acc[r] * (1.0f / CDIM);
        }
    }
    __syncthreads();

    // Exact top-k membership: rank_i = #{s_j > s_i} + #{j<i : s_j == s_i}.
    const float si = s_scores[tid];
    int cnt = 0;
    for (int j = 0; j < HWDIM; ++j) {
        const float sj = s_scores[j];              // uniform j -> LDS broadcast
        cnt += (sj > si) || ((sj == si) && (j < tid));
    }
    mask_out[(size_t)bt * HWDIM + tid] = (cnt < KEEP) ? 1.0f : 0.0f;
}

// ---------------------------------------------------------------------------
// Kernel B: pruned = x * mask, float4-wide. x hits L2 (warmed by kernel A,
// 33.5MB << 192MB L2); mask region is 128KB and stays L2-hot; stores are
// non-temporal (never re-read).
// ---------------------------------------------------------------------------
__global__ __launch_bounds__(256)
void tokenprune_apply(const float* __restrict__ x,
                      const float* __restrict__ mask,
                      float* __restrict__ out,
                      int n4)
{
    const int i4 = blockIdx.x * 256 + threadIdx.x;
    if (i4 >= n4) return;
    const size_t flat = (size_t)i4 * 4;
    const int bt = (int)(flat >> 18);              // / (CDIM*HWDIM)
    const int p  = (int)(flat & (HWDIM - 1));      // position (4-aligned)

    const f32x4 xv = *(const f32x4*)(x + flat);
    const f32x4 mv = *(const f32x4*)(mask + (size_t)bt * HWDIM + p);
    const f32x4 ov = xv * mv;
    __builtin_nontemporal_store(ov, (f32x4*)(out + flat));
}

extern "C" void kernel_launch(void* const* d_in, const int* in_sizes, int n_in,
                              void* d_out, int out_size, void* d_ws, size_t ws_size,
                              hipStream_t stream)
{
    (void)in_sizes; (void)n_in; (void)d_ws; (void)ws_size; (void)out_size;

    const float* x  = (const float*)d_in[0];
    float* out      = (float*)d_out;
    float* mask_out = out + NPRUNED;               // mask tail of d_out (float 0/1)

    // Pass A: scores (WMMA) + top-k mask. 32 blocks x 1024 threads.
    tokenprune_score_select<<<NBT, 1024, 0, stream>>>(x, mask_out);

    // Pass B: streaming apply. 8.4M f32 = 2.1M float4.
    const int n4 = (int)(NPRUNED / 4);
    tokenprune_apply<<<(n4 + 255) / 256, 256, 0, stream>>>(x, mask_out, out, n4);
}